// LocallyConnected1d_15668040696319
// MI455X (gfx1250) — compile-verified
//
#include <hip/hip_runtime.h>

// LocallyConnected1d: out[b,o,l] = sum_{i,k} x[b,i,l+k] * w[o,i,l,k]
// Per-l GEMM (M=32 batch, N=32 out-ch, K=224=i*7+k) on the CDNA5 matrix
// pipe via V_WMMA_F32_16X16X4_F32 (fp32 in/out, matches reference precision).
// Memory-bound on the 117MB weight stream -> stage per-l operand slices in
// LDS with contiguous 7-float runs per (row,channel) pair so the weight
// stream is consumed cacheline-efficiently across the LT consecutive l's
// handled by each workgroup.

typedef __attribute__((ext_vector_type(2))) float v2f;
typedef __attribute__((ext_vector_type(8))) float v8f;

#define B_    32
#define IC    32
#define L_    4096
#define OC    32
#define KW    7
#define LOUT  4090
#define KKTOT (IC * KW)   // 224 contraction values per l
#define LT    4           // l's per workgroup
#define ROW   33          // LDS row stride (pad to de-align banks)

__global__ __launch_bounds__(128, 2)
void lc1d_wmma(const float* __restrict__ x,
               const float* __restrict__ w,
               float* __restrict__ out) {
  // [kk][col] slices for the current l: col = b for xs, col = o for ws.
  __shared__ float xs[KKTOT * ROW];   // 29,568 B
  __shared__ float ws[KKTOT * ROW];   // 29,568 B

  const int tid  = threadIdx.x;       // 0..127
  const int lane = tid & 31;
  const int wave = tid >> 5;          // 0..3
  const int bh   = wave & 1;          // which 16-row batch half
  const int oh   = wave >> 1;         // which 16-col out-channel half
  const int m    = lane & 15;         // row/col within 16x16 tile
  const int half = lane >> 4;         // K-pair select within WMMA frag

  const int l0 = blockIdx.x * LT;

  for (int lt = 0; lt < LT; ++lt) {
    const int l = l0 + lt;            // uniform across the workgroup
    if (l >= LOUT) break;             // uniform tail guard: EXEC stays full

    __syncthreads();                  // previous iteration's readers done
    // Cooperative stage: 1024 (col,i) pairs, 7 contiguous floats each.
    for (int p = tid; p < B_ * IC; p += 128) {
      const int col = p >> 5;         // b for x, o for w (both 0..31)
      const int i   = p & 31;
      const float* xsrc = x + ((size_t)col * IC + i) * L_ + l;          // x[b][i][l..l+6]
      const float* wsrc = w + (((size_t)col * IC + i) * LOUT + l) * KW; // w[o][i][l][0..6]
      const int rowb = i * KW;
      #pragma unroll
      for (int k = 0; k < KW; ++k) {
        xs[(rowb + k) * ROW + col] = xsrc[k];
        ws[(rowb + k) * ROW + col] = wsrc[k];
      }
    }
    __syncthreads();

    // 56 x V_WMMA_F32_16X16X4_F32 accumulating C[16b x 16o].
    // A frag (16x4 f32): v0 = K at (4g + 2*half), v1 = next K.
    // B frag (4x16 f32): same K rows, lane = column.
    v8f acc = {};
    const float* xbase = xs + (2 * half) * ROW + bh * 16 + m;
    const float* wbase = ws + (2 * half) * ROW + oh * 16 + m;
    #pragma unroll 8
    for (int g = 0; g < KKTOT / 4; ++g) {
      v2f a, bm;
      a.x  = xbase[(g * 4) * ROW];
      a.y  = xbase[(g * 4) * ROW + ROW];
      bm.x = wbase[(g * 4) * ROW];
      bm.y = wbase[(g * 4) * ROW + ROW];
      acc = __builtin_amdgcn_wmma_f32_16x16x4_f32(
          /*neg_a=*/false, a, /*neg_b=*/false, bm,
          /*c_mod=*/(short)0, acc, /*reuse_a=*/false, /*reuse_b=*/false);
    }

    // D layout: VGPR r -> M = 8*half + r (lanes split), N = m.
    const int ocol = oh * 16 + m;
    #pragma unroll
    for (int r = 0; r < 8; ++r) {
      const int brow = bh * 16 + half * 8 + r;
      out[((size_t)brow * OC + ocol) * LOUT + l] = acc[r];
    }
  }
}

extern "C" void kernel_launch(void* const* d_in, const int* in_sizes, int n_in,
                              void* d_out, int out_size, void* d_ws, size_t ws_size,
                              hipStream_t stream) {
  const float* x = (const float*)d_in[0];   // (32, 32, 4096) f32
  const float* w = (const float*)d_in[1];   // (32, 32, 4090, 7) f32
  float* out = (float*)d_out;               // (32, 32, 4090) f32

  const int grid = (LOUT + LT - 1) / LT;    // 1023 workgroups
  lc1d_wmma<<<grid, 128, 0, stream>>>(x, w, out);
}